// StructureMeasureLoss_42434276884739
// MI455X (gfx1250) — compile-verified
//
#include <hip/hip_runtime.h>

typedef __attribute__((ext_vector_type(2))) float v2f;
typedef __attribute__((ext_vector_type(8))) float v8f;

#define IMG   512
#define OUT_D 502      // 512 - 11 + 1
#define C3F   0.0001f
#define HSTRIDE 34     // H tile column stride in LDS (even + conflict-spreading)

// D = A(16x4, f32) * B(4x16, f32) + C(16x16, f32)
__device__ __forceinline__ v8f wmma4(v2f a, v2f b, v8f c) {
    return __builtin_amdgcn_wmma_f32_16x16x4_f32(
        /*neg_a=*/false, a, /*neg_b=*/false, b,
        /*c_mod=*/(short)0, c, /*reuse_a=*/false, /*reuse_b=*/false);
}

__global__ void __launch_bounds__(128)
ssim_wmma_kernel(const float* __restrict__ I1, const float* __restrict__ I2,
                 const float* __restrict__ win, double* __restrict__ acc)
{
    __shared__ float gpad[64];              // zero-padded taps: gpad[16+d]=g[d], d in [0,11)
    __shared__ float hLds[4 * 16 * HSTRIDE]; // per-wave H tile, column-major
    __shared__ float redLds[128];

    const int tid  = threadIdx.x;
    const int lane = tid & 31;
    const int wave = tid >> 5;
    const int m    = lane & 15;       // M row (pass2) / N col (pass1 B, pass2 B)
    const int hi   = lane >> 4;       // lane half
    const int koff0 = hi * 2;         // K offset of fragment .x element

    // zero-padded 1D gaussian: g[i] = w2d[i][5] / sqrt(w2d[5][5])
    if (tid < 64) {
        float v = 0.0f;
        int d = tid - 16;
        if ((unsigned)d < 11u) v = win[d * 11 + 5] * rsqrtf(win[5 * 11 + 5]);
        gpad[tid] = v;
    }
    __syncthreads();

    // Banded-weight fragments, shared by pass1-B and pass2-A:
    // wf[s].{x,y} = g[(4s+koff0+{0,1}) - m]   (unconditional padded-LDS loads)
    v2f wf[8];
#pragma unroll
    for (int s = 0; s < 8; ++s) {
        int idx = 16 + 4 * s + koff0 - m;     // in [1,47] -> always in-bounds
        wf[s].x = gpad[idx];
        wf[s].y = gpad[idx + 1];
    }

    // One 16x16 output tile per wave; 32x32 tiles per image (edges masked).
    int waveGlobal = blockIdx.x * 4 + wave;
    int b   = waveGlobal >> 10;
    int rem = waveGlobal & 1023;
    int ty  = rem >> 5, tx = rem & 31;
    int r0  = ty * 16, c0 = tx * 16;

    const float* p1 = I1 + (size_t)b * IMG * IMG;
    const float* p2 = I2 + (size_t)b * IMG * IMG;
    float* hw = &hLds[wave * 16 * HSTRIDE];

    // ---- register-cache the 32x32 halo of both inputs (v2f pairs) ----
    v2f x1c[2][8], x2c[2][8];
    bool interior = (ty < 31) && (tx < 31);   // wave-uniform
    if (interior) {
#pragma unroll
        for (int t = 0; t < 2; ++t) {
            const float* row1 = p1 + (size_t)(r0 + 16 * t + m) * IMG + c0;
            const float* row2 = p2 + (size_t)(r0 + 16 * t + m) * IMG + c0;
#pragma unroll
            for (int s = 0; s < 8; ++s) {
                int k0 = 4 * s + koff0;       // even -> 8B-aligned
                x1c[t][s] = *(const v2f*)(row1 + k0);
                x2c[t][s] = *(const v2f*)(row2 + k0);
            }
        }
    } else {
#pragma unroll
        for (int t = 0; t < 2; ++t) {
            int gr = r0 + 16 * t + m;
            const float* row1 = p1 + (size_t)gr * IMG;
            const float* row2 = p2 + (size_t)gr * IMG;
#pragma unroll
            for (int s = 0; s < 8; ++s) {
                int gc = c0 + 4 * s + koff0;
                v2f a = {0.f, 0.f}, bb = {0.f, 0.f};
                if (gr < IMG) {
                    if (gc < IMG)     { a.x = row1[gc];     bb.x = row2[gc]; }
                    if (gc + 1 < IMG) { a.y = row1[gc + 1]; bb.y = row2[gc + 1]; }
                }
                x1c[t][s] = a;
                x2c[t][s] = bb;
            }
        }
    }

    v8f accO[5];

#pragma unroll
    for (int ch = 0; ch < 5; ++ch) {
        // ---- pass 1: H(32x16) = X(32x32) * Wh(32x16) ----
        v8f hAcc0 = {0.f,0.f,0.f,0.f,0.f,0.f,0.f,0.f};
        v8f hAcc1 = {0.f,0.f,0.f,0.f,0.f,0.f,0.f,0.f};
#pragma unroll
        for (int t = 0; t < 2; ++t) {
#pragma unroll
            for (int s = 0; s < 8; ++s) {
                v2f xa = x1c[t][s], xb = x2c[t][s];
                v2f A;
                switch (ch) {
                    case 0:  A = xa; break;
                    case 1:  A = xb; break;
                    case 2:  A.x = xa.x * xa.x; A.y = xa.y * xa.y; break;
                    case 3:  A.x = xb.x * xb.x; A.y = xb.y * xb.y; break;
                    default: A.x = xa.x * xb.x; A.y = xa.y * xb.y; break;
                }
                if (t == 0) hAcc0 = wmma4(A, wf[s], hAcc0);
                else        hAcc1 = wmma4(A, wf[s], hAcc1);
            }
        }

        // ---- spill H to LDS, column-major (col m, row rr) ----
#pragma unroll
        for (int i = 0; i < 8; ++i) {
            hw[m * HSTRIDE + (i + hi * 8)]       = hAcc0[i];
            hw[m * HSTRIDE + (16 + i + hi * 8)]  = hAcc1[i];
        }
        __syncthreads();

        // ---- pass 2: O(16x16) = Wv(16x32) * H(32x16); A2 == wf ----
        v8f o = {0.f,0.f,0.f,0.f,0.f,0.f,0.f,0.f};
#pragma unroll
        for (int s = 0; s < 8; ++s) {
            int k0 = 4 * s + koff0;                       // even -> 8B-aligned
            v2f B = *(const v2f*)&hw[m * HSTRIDE + k0];   // ds_load_b64
            o = wmma4(wf[s], B, o);
        }
        accO[ch] = o;
        __syncthreads();
    }

    // ---- pointwise SSIM + masked accumulation ----
    float lsum = 0.0f;
#pragma unroll
    for (int i = 0; i < 8; ++i) {
        int orow = r0 + i + hi * 8;
        int ocol = c0 + m;
        if (orow < OUT_D && ocol < OUT_D) {
            float mu1 = accO[0][i], mu2 = accO[1][i];
            float e11 = accO[2][i], e22 = accO[3][i], e12 = accO[4][i];
            float s1  = e11 - mu1 * mu1;
            float s2  = e22 - mu2 * mu2;
            float s12 = e12 - mu1 * mu2;
            lsum += (s12 + C3F) / (sqrtf(fabsf(s1)) * sqrtf(fabsf(s2)) + C3F);
        }
    }

    redLds[tid] = lsum;
    __syncthreads();
    if (tid == 0) {
        float t = 0.0f;
        for (int i = 0; i < 128; ++i) t += redLds[i];
        atomicAdd(acc, (double)t);
    }
}

__global__ void zero_acc_kernel(double* acc) { acc[0] = 0.0; }

__global__ void finalize_kernel(const double* acc, float* out) {
    out[0] = (float)(acc[0] * (1.0 / (32.0 * 502.0 * 502.0)));
}

extern "C" void kernel_launch(void* const* d_in, const int* in_sizes, int n_in,
                              void* d_out, int out_size, void* d_ws, size_t ws_size,
                              hipStream_t stream) {
    const float* input1 = (const float*)d_in[0];
    const float* input2 = (const float*)d_in[1];
    const float* window = (const float*)d_in[2];
    float* out = (float*)d_out;
    double* acc = (double*)d_ws;

    zero_acc_kernel<<<1, 1, 0, stream>>>(acc);
    // 32 images * 32*32 tiles = 32768 tiles; 4 waves (tiles) per 128-thread block
    ssim_wmma_kernel<<<8192, 128, 0, stream>>>(input1, input2, window, acc);
    finalize_kernel<<<1, 1, 0, stream>>>(acc, out);
}